// CLS_4604204942081
// MI455X (gfx1250) — compile-verified
//
#include <hip/hip_runtime.h>
#include <math.h>

#define N_NODES 100000
#define N_EDGES 1600000
#define IN_SIZE 256
#define OUT_SIZE 40

typedef __attribute__((ext_vector_type(2))) float v2f;
typedef __attribute__((ext_vector_type(8))) float v8f;

// ---------------------------------------------------------------------------
// 0) Init workspace: deg = 1.0 (self-loop), accum = 0
// ---------------------------------------------------------------------------
__global__ void gcn_init(float* __restrict__ deg, float* __restrict__ accum) {
    int t = blockIdx.x * blockDim.x + threadIdx.x;
    if (t < N_NODES * OUT_SIZE) accum[t] = 0.0f;
    if (t < N_NODES) deg[t] = 1.0f;  // self-loop contribution to degree
}

// ---------------------------------------------------------------------------
// 1) In-degree via f32 atomics (L2-resident, 400 KB)
// ---------------------------------------------------------------------------
__global__ void gcn_degree(const int* __restrict__ dst, float* __restrict__ deg) {
    int e = blockIdx.x * blockDim.x + threadIdx.x;
    if (e < N_EDGES) atomicAdd(&deg[dst[e]], 1.0f);
}

// ---------------------------------------------------------------------------
// 2) deg -> rsqrt(deg) in place (deg >= 1 always, so no zero guard needed)
// ---------------------------------------------------------------------------
__global__ void gcn_rsqrt(float* __restrict__ deg) {
    int i = blockIdx.x * blockDim.x + threadIdx.x;
    if (i < N_NODES) deg[i] = rsqrtf(deg[i]);
}

// ---------------------------------------------------------------------------
// 3) h = x @ W via V_WMMA_F32_16X16X4_F32, full fp32 precision.
//    One wave computes a 16-row x 48-col tile (3 WMMA N-tiles; cols 40..47
//    masked to zero). K loop: 256/4 = 64 WMMA steps per N-tile.
//    A 16x4 f32 layout: lanes 0-15 hold K = {0,1}, lanes 16-31 hold K = {2,3}.
//    B 4x16 f32 layout mirrors A: lane = column, VGPR+halfwave = K row.
//    C/D 16x16 f32: VGPR r <-> row r + 8*half, lane&15 = column.
// ---------------------------------------------------------------------------
__global__ void __launch_bounds__(128) gcn_gemm_wmma(const float* __restrict__ x,
                                                     const float* __restrict__ W,
                                                     float* __restrict__ h) {
    const int lane  = threadIdx.x & 31;
    const int wave  = threadIdx.x >> 5;
    const int mtile = blockIdx.x * 4 + wave;
    if (mtile >= N_NODES / 16) return;   // wave-uniform exit: EXEC stays full

    const int m0   = mtile * 16;
    const int half = lane >> 4;          // 0: K={0,1}, 1: K={2,3}
    const int l16  = lane & 15;

    const float* aptr = x + (size_t)(m0 + l16) * IN_SIZE + half * 2;
    const int   c2col = 32 + (l16 & 7);               // clamp tile-2 column
    const float mask2 = (l16 < 8) ? 1.0f : 0.0f;      // zero cols 40..47

    v8f c0 = {}; v8f c1 = {}; v8f c2 = {};

    for (int k = 0; k < IN_SIZE; k += 4) {
        v2f a = *(const v2f*)(aptr + k);              // 8B load: K = k+2*half, +1

        const float* w0 = W + (size_t)(k + half * 2) * OUT_SIZE;
        const float* w1 = w0 + OUT_SIZE;
        v2f b0 = { w0[l16],            w1[l16]            };
        v2f b1 = { w0[16 + l16],       w1[16 + l16]       };
        v2f b2 = { w0[c2col] * mask2,  w1[c2col] * mask2  };

        c0 = __builtin_amdgcn_wmma_f32_16x16x4_f32(false, a, false, b0,
                                                   (short)0, c0, false, false);
        c1 = __builtin_amdgcn_wmma_f32_16x16x4_f32(false, a, false, b1,
                                                   (short)0, c1, false, false);
        c2 = __builtin_amdgcn_wmma_f32_16x16x4_f32(false, a, false, b2,
                                                   (short)0, c2, false, false);
    }

    // Store D: VGPR r holds row m0 + 8*half + r, column n0 + l16
    for (int r = 0; r < 8; ++r) {
        float* hr = h + (size_t)(m0 + half * 8 + r) * OUT_SIZE;
        hr[l16]      = c0[r];
        hr[16 + l16] = c1[r];
        if (l16 < 8) hr[32 + l16] = c2[r];
    }
}

// ---------------------------------------------------------------------------
// 4) Edge scatter: each lane owns one edge; wave broadcasts (src,dst,norm)
//    per edge so the 40-float gather + 40 atomics are lane-coalesced.
//    accum is 16 MB -> lives in the 192 MB L2, atomics stay on-chip.
// ---------------------------------------------------------------------------
__global__ void gcn_scatter(const float* __restrict__ h,
                            const float* __restrict__ dinv,
                            const int* __restrict__ src,
                            const int* __restrict__ dst,
                            float* __restrict__ accum) {
    int e    = blockIdx.x * blockDim.x + threadIdx.x;
    int lane = threadIdx.x & 31;

    int s = -1, d = -1;
    float w = 0.0f;
    if (e < N_EDGES) {
        s = src[e];
        d = dst[e];
        w = dinv[s] * dinv[d];
    }

    for (int p = 0; p < 32; ++p) {
        int sp = __shfl(s, p, 32);     // wave-uniform after broadcast
        if (sp < 0) continue;
        int   dp = __shfl(d, p, 32);
        float wp = __shfl(w, p, 32);

        atomicAdd(&accum[(size_t)dp * OUT_SIZE + lane],
                  h[(size_t)sp * OUT_SIZE + lane] * wp);
        if (lane < 8)
            atomicAdd(&accum[(size_t)dp * OUT_SIZE + 32 + lane],
                      h[(size_t)sp * OUT_SIZE + 32 + lane] * wp);
    }
}

// ---------------------------------------------------------------------------
// 5) Fused self-loop + bias + log_softmax. One wave per row (wave32):
//    lane covers col j and j+32 (j<8); shfl_xor tree reductions for max/sum.
// ---------------------------------------------------------------------------
__global__ void gcn_finalize(const float* __restrict__ accum,
                             const float* __restrict__ h,
                             const float* __restrict__ dinv,
                             const float* __restrict__ b,
                             float* __restrict__ out) {
    int lane = threadIdx.x & 31;
    int wave = threadIdx.x >> 5;
    int row  = blockIdx.x * 8 + wave;    // blockDim = 256 -> 8 rows/block
    if (row >= N_NODES) return;

    float dii = dinv[row];
    float sl  = dii * dii;               // self-loop norm = dinv[i]*dinv[i]
    size_t base = (size_t)row * OUT_SIZE;

    float v0 = accum[base + lane] + h[base + lane] * sl + b[lane];
    float v1 = -INFINITY;
    if (lane < 8)
        v1 = accum[base + 32 + lane] + h[base + 32 + lane] * sl + b[32 + lane];

    float m = fmaxf(v0, v1);
    for (int off = 16; off > 0; off >>= 1)
        m = fmaxf(m, __shfl_xor(m, off, 32));

    float e0 = expf(v0 - m);
    float e1 = (lane < 8) ? expf(v1 - m) : 0.0f;
    float s  = e0 + e1;
    for (int off = 16; off > 0; off >>= 1)
        s += __shfl_xor(s, off, 32);

    float l = logf(s) + m;
    out[base + lane] = v0 - l;
    if (lane < 8) out[base + 32 + lane] = v1 - l;
}

// ---------------------------------------------------------------------------
extern "C" void kernel_launch(void* const* d_in, const int* in_sizes, int n_in,
                              void* d_out, int out_size, void* d_ws, size_t ws_size,
                              hipStream_t stream) {
    const float* x    = (const float*)d_in[0];   // [N, 256]
    const float* W    = (const float*)d_in[1];   // [256, 40]
    const float* b    = (const float*)d_in[2];   // [40]
    const int*   edge = (const int*)d_in[3];     // [2, E] flattened
    const int*   src  = edge;
    const int*   dst  = edge + N_EDGES;

    float* ws    = (float*)d_ws;
    float* deg   = ws;                                    // N      (deg -> dinv)
    float* h     = ws + N_NODES;                          // N * 40
    float* accum = h + (size_t)N_NODES * OUT_SIZE;        // N * 40
    float* out   = (float*)d_out;

    const int mtiles = N_NODES / 16;                      // 6250

    gcn_init    <<<(N_NODES * OUT_SIZE + 255) / 256, 256, 0, stream>>>(deg, accum);
    gcn_degree  <<<(N_EDGES + 255) / 256,            256, 0, stream>>>(dst, deg);
    gcn_rsqrt   <<<(N_NODES + 255) / 256,            256, 0, stream>>>(deg);
    gcn_gemm_wmma<<<(mtiles + 3) / 4,                128, 0, stream>>>(x, W, h);
    gcn_scatter <<<(N_EDGES + 255) / 256,            256, 0, stream>>>(h, deg, src, dst, accum);
    gcn_finalize<<<(N_NODES + 7) / 8,                256, 0, stream>>>(accum, h, deg, b, out);
}